// DGCL_Loss_2259152797809
// MI455X (gfx1250) — compile-verified
//
#include <hip/hip_runtime.h>

typedef __attribute__((ext_vector_type(2))) float v2f;
typedef __attribute__((ext_vector_type(8))) float v8f;
typedef __attribute__((ext_vector_type(4))) unsigned int v4u;
typedef __attribute__((ext_vector_type(8))) int v8i;
typedef __attribute__((ext_vector_type(4))) int v4i;

#define BSZ 4096
#define DIM 512
#define NN 1000000.0f
#define TEMP 0.07f
#define INV_T (1.0f / TEMP)
#define GAMMA 0.9f
#define THETA 0.9f
#define ETA_INIT 0.01f
#define START_EPOCHS 5
#define EPS_CLAMP 1e-16f
#define PI_F 3.14159265358979323846f

// GEMM blocking
#define TM 128  // M rows per block
#define TN 64   // N cols per block
#define KC 64   // K chunk staged in LDS
#define LDA 68  // padded LDS row stride (floats): 68*4B = 272B, 16B aligned,
#define LDB 68  // 4-bank skew per row -> conflict-free 16-lane frag reads

#define SIM_FLOATS (BSZ * BSZ)
// scratch arrays after sim, each 4096 floats:
// 0:diag 1:rmax_sim 2:cmax_dtxt 3:nb_I 4:nb_T 5:old_b_I 6:old_b_T
// 7:zeta_I_b 8:zeta_T_b 9:rowterm(diag+nb_I) 10:colterm(diag+nb_T)
// 11:g_I 12:num_img 13:recipI_tilde 14:image_loss
// 15:g_T 16:num_txt 17:recipT_tilde 18:text_loss
// 19:colsum_wimg_tilde 20:rowsum_wtxt_tilde 21:zetaI_new 22:zetaT_new
// 23:[0]=cur_eta
#define ARR(k) (ws + SIM_FLOATS + (k) * BSZ)

#if defined(__gfx1250__) && __has_builtin(__builtin_amdgcn_tensor_load_to_lds)
#define USE_TDM 1
#endif

// ---------------------------------------------------------------------------
// TDM: DMA a [tile_rows x KC] f32 tile (row stride 512 elements in memory)
// into LDS with 4-DWORD padding every 64 DWORDs (=> 68-float LDS row stride).
// D# layout per CDNA5 ISA ch.8 (group0: count/lds_addr/global_addr/type=2,
// group1: data_size=4B, pad_enable, pad_interval code 5 (64 dw),
// pad_amount code 3 (4 dw), tensor dims/strides, tile dims).
// ---------------------------------------------------------------------------
#ifdef USE_TDM
__device__ __forceinline__ void tdm_load_tile(unsigned lds_off,
                                              const float* gptr,
                                              unsigned tile_rows) {
  unsigned long long ga = (unsigned long long)gptr;
  v4u g0;
  g0[0] = 1u;                                                // count=1, user D#
  g0[1] = lds_off;                                           // lds_addr (bytes)
  g0[2] = (unsigned)(ga & 0xffffffffu);                      // global_addr lo
  g0[3] = (unsigned)((ga >> 32) & 0x01ffffffu) | (2u << 30); // addr hi | type=2
  v8i g1;
  g1[0] = (int)((2u << 16) |      // data_size = 4 bytes
                (1u << 20) |      // pad_enable
                (5u << 22) |      // pad_interval: 64 DWORDs
                (3u << 25));      // pad_amount: 4 DWORDs
  g1[1] = (int)(512u << 16);      // tensor_dim0 = 512 (lo 16 bits)
  g1[2] = (int)(4096u << 16);     // tensor_dim0 hi=0 | tensor_dim1 = 4096 (lo)
  g1[3] = (int)((unsigned)KC << 16);  // tensor_dim1 hi=0 | tile_dim0 = KC
  g1[4] = (int)tile_rows;         // tile_dim1 = rows | tile_dim2 = 0
  g1[5] = 512;                    // tensor_dim0_stride = 512 (lo 32)
  g1[6] = 0;                      // stride0 hi | stride1 lo
  g1[7] = 0;
  v4i gz = {0, 0, 0, 0};
#if __clang_major__ >= 23
  v8i gz8 = {0, 0, 0, 0, 0, 0, 0, 0};
  __builtin_amdgcn_tensor_load_to_lds(g0, g1, gz, gz, gz8, 0);
#else
  __builtin_amdgcn_tensor_load_to_lds(g0, g1, gz, gz, 0);
#endif
}
#endif

// ---------------------------------------------------------------------------
// Kernel 1: sim = img @ txt^T via V_WMMA_F32_16X16X4_F32.
// Block = 256 threads (8 waves) -> 128x64 output tile; wave (4x2 grid) owns a
// 32x32 tile = 4 accumulators, so each k-step is 4 ds_load_b64 -> 4 WMMAs.
// A/B K-chunks staged in LDS by the Tensor Data Mover (wave 0 issues,
// s_wait_tensorcnt 0, workgroup barrier); cooperative-load fallback otherwise.
// A 16x4 f32 frag layout: lanes 0-15 M=lane K0/K1, lanes 16-31 K2/K3.
// ---------------------------------------------------------------------------
__global__ void k_gemm(const float* __restrict__ img,
                       const float* __restrict__ txt,
                       float* __restrict__ sim) {
  __shared__ float As[TM * LDA];
  __shared__ float Bs[TN * LDB];
  const int lane = threadIdx.x & 31;
  const int wave = threadIdx.x >> 5;
  const int waveM = wave >> 1;  // 0..3
  const int waveN = wave & 1;   // 0..1
  const int r = lane & 15;
  const int hi = lane >> 4;     // 0: K=0,1   1: K=2,3
  const int gm = blockIdx.y * TM;
  const int gn = blockIdx.x * TN;

  v8f c[2][2];
#pragma unroll
  for (int mi = 0; mi < 2; ++mi)
#pragma unroll
    for (int ni = 0; ni < 2; ++ni)
      c[mi][ni] = (v8f){0.f, 0.f, 0.f, 0.f, 0.f, 0.f, 0.f, 0.f};

  for (int kc = 0; kc < DIM; kc += KC) {
    __syncthreads();  // previous chunk's LDS reads complete
#ifdef USE_TDM
    if (wave == 0) {
      tdm_load_tile((unsigned)(unsigned long long)(uintptr_t)As,
                    img + (size_t)gm * DIM + kc, TM);
      tdm_load_tile((unsigned)(unsigned long long)(uintptr_t)Bs,
                    txt + (size_t)gn * DIM + kc, TN);
      __builtin_amdgcn_s_wait_tensorcnt(0);
    }
#else
    for (int idx = threadIdx.x; idx < TM * (KC / 4); idx += 256) {
      const int row = idx >> 4, c4 = idx & 15;
      *(float4*)(&As[row * LDA + c4 * 4]) =
          *(const float4*)(img + (size_t)(gm + row) * DIM + kc + c4 * 4);
    }
    for (int idx = threadIdx.x; idx < TN * (KC / 4); idx += 256) {
      const int row = idx >> 4, c4 = idx & 15;
      *(float4*)(&Bs[row * LDB + c4 * 4]) =
          *(const float4*)(txt + (size_t)(gn + row) * DIM + kc + c4 * 4);
    }
#endif
    __syncthreads();
#pragma unroll 4
    for (int k = 0; k < KC; k += 4) {
      const int kk = k + hi * 2;
      const v2f a0 = *(const v2f*)(&As[(waveM * 32 + r) * LDA + kk]);
      const v2f a1 = *(const v2f*)(&As[(waveM * 32 + 16 + r) * LDA + kk]);
      const v2f b0 = *(const v2f*)(&Bs[(waveN * 32 + r) * LDB + kk]);
      const v2f b1 = *(const v2f*)(&Bs[(waveN * 32 + 16 + r) * LDB + kk]);
      c[0][0] = __builtin_amdgcn_wmma_f32_16x16x4_f32(false, a0, false, b0,
                                                      (short)0, c[0][0], false, false);
      c[0][1] = __builtin_amdgcn_wmma_f32_16x16x4_f32(false, a0, false, b1,
                                                      (short)0, c[0][1], false, false);
      c[1][0] = __builtin_amdgcn_wmma_f32_16x16x4_f32(false, a1, false, b0,
                                                      (short)0, c[1][0], false, false);
      c[1][1] = __builtin_amdgcn_wmma_f32_16x16x4_f32(false, a1, false, b1,
                                                      (short)0, c[1][1], false, false);
    }
  }

  // C/D layout: VGPR v holds M = v + 8*hi, N = r
#pragma unroll
  for (int mi = 0; mi < 2; ++mi) {
#pragma unroll
    for (int ni = 0; ni < 2; ++ni) {
      const int row0 = gm + waveM * 32 + mi * 16 + hi * 8;
      const int col = gn + waveN * 32 + ni * 16 + r;
#pragma unroll
      for (int v = 0; v < 8; ++v)
        sim[(size_t)(row0 + v) * BSZ + col] = c[mi][ni][v];
    }
  }
}

// ---------------------------------------------------------------------------
// Kernel 2: per-row max of sim and diagonal extraction
// ---------------------------------------------------------------------------
__global__ void k_rowmax(const float* __restrict__ sim, float* __restrict__ ws) {
  __shared__ float sm[256];
  const int i = blockIdx.x;
  const float* row = sim + (size_t)i * BSZ;
  float m = -INFINITY;
  for (int j = threadIdx.x; j < BSZ; j += 256) m = fmaxf(m, row[j]);
  sm[threadIdx.x] = m;
  __syncthreads();
  for (int s = 128; s > 0; s >>= 1) {
    if (threadIdx.x < s) sm[threadIdx.x] = fmaxf(sm[threadIdx.x], sm[threadIdx.x + s]);
    __syncthreads();
  }
  if (threadIdx.x == 0) {
    ARR(1)[i] = sm[0];
    ARR(0)[i] = row[i];
  }
}

// ---------------------------------------------------------------------------
// Kernel 3: per-column max of d_txt = sim - diag[j]; 64 cols/block, coalesced
// ---------------------------------------------------------------------------
__global__ void k_colmax(const float* __restrict__ sim, float* __restrict__ ws) {
  __shared__ float sm[256];
  const int tc = threadIdx.x & 63, tr = threadIdx.x >> 6;
  const int j = blockIdx.x * 64 + tc;
  float m = -INFINITY;
  for (int i = tr; i < BSZ; i += 4) m = fmaxf(m, sim[(size_t)i * BSZ + j]);
  sm[threadIdx.x] = m;
  __syncthreads();
  if (tr == 0) {
    m = fmaxf(fmaxf(sm[tc], sm[tc + 64]), fmaxf(sm[tc + 128], sm[tc + 192]));
    ARR(2)[j] = m - ARR(0)[j];
  }
}

// ---------------------------------------------------------------------------
// Kernel 4: gather id-indexed state, nb_I/nb_T, row/col terms, cur_eta
// ---------------------------------------------------------------------------
__global__ void k_prep(const int* __restrict__ idI, const int* __restrict__ idT,
                       const float* __restrict__ b_I, const float* __restrict__ b_T,
                       const float* __restrict__ zeta_I, const float* __restrict__ zeta_T,
                       const int* __restrict__ epoch, const int* __restrict__ max_epoch,
                       float* __restrict__ ws) {
  const int i = blockIdx.x * 256 + threadIdx.x;
  if (i < BSZ) {
    const int ii = idI[i], it = idT[i];
    const float obI = b_I[ii], obT = b_T[it];
    const float di = ARR(0)[i];
    const float nbI = fmaxf(ARR(1)[i] - di, obI);
    const float nbT = fmaxf(ARR(2)[i], obT);
    ARR(3)[i] = nbI;
    ARR(4)[i] = nbT;
    ARR(5)[i] = obI;
    ARR(6)[i] = obT;
    ARR(7)[i] = zeta_I[ii];
    ARR(8)[i] = zeta_T[it];
    ARR(9)[i] = di + nbI;
    ARR(10)[i] = di + nbT;
  }
  if (i == 0) {
    const int ep = epoch[0], me = max_epoch[0];
    float eta = 0.f;
    if (ep >= START_EPOCHS) {
      float base = 0.5f * ETA_INIT *
                   (1.f + cosf(PI_F * (float)(ep - START_EPOCHS) /
                               (float)(me - 1 - START_EPOCHS)));
      if (ep < me / 2) eta = base;
      else if (ep < (me * 3) / 4) eta = base * 0.1f;
      else eta = base * 0.01f;
    }
    ARR(23)[0] = eta;
  }
}

// ---------------------------------------------------------------------------
// Kernel 5: row pass -> g_I[i], num_img[i]
// ---------------------------------------------------------------------------
__global__ void k_rowpass1(const float* __restrict__ sim, float* __restrict__ ws) {
  __shared__ float sg[256], sn[256];
  const int i = blockIdx.x;
  const float* row = sim + (size_t)i * BSZ;
  const float* zTb = ARR(8);
  const float di = ARR(0)[i];
  const float rt = ARR(9)[i];  // diag[i] + nb_I[i]
  float g = 0.f, num = 0.f;
  for (int j = threadIdx.x; j < BSZ; j += 256) {
    const float s = row[j];
    const float e = expf((s - rt - zTb[j]) * INV_T);
    if (j != i) {
      g += e;
      num += e * (s - di);
    }
  }
  sg[threadIdx.x] = g;
  sn[threadIdx.x] = num;
  __syncthreads();
  for (int s = 128; s > 0; s >>= 1) {
    if (threadIdx.x < s) {
      sg[threadIdx.x] += sg[threadIdx.x + s];
      sn[threadIdx.x] += sn[threadIdx.x + s];
    }
    __syncthreads();
  }
  if (threadIdx.x == 0) {
    ARR(11)[i] = sg[0];
    ARR(12)[i] = sn[0];
  }
}

// ---------------------------------------------------------------------------
// Kernel 6: s_I moving average, image_loss, 1/denomI_tilde
// ---------------------------------------------------------------------------
__global__ void k_smallI(const float* __restrict__ s_I, const int* __restrict__ idI,
                         float* __restrict__ ws) {
  const int i = blockIdx.x * 256 + threadIdx.x;
  if (i >= BSZ) return;
  const float g = ARR(11)[i];
  const float obI = ARR(5)[i], nbI = ARR(3)[i];
  float s0 = s_I[idI[i]];
  if (s0 == 0.f) s0 = g * expf(nbI - obI);
  const float sI = (1.f - GAMMA) * s0 * expf(obI - nbI) + GAMMA * g;
  const float offI = ((float)BSZ - 1.f) / (NN - 1.f) * expf(-nbI * INV_T);
  const float denom = fmaxf(sI + offI, EPS_CLAMP);
  ARR(14)[i] = ARR(12)[i] / denom;  // image_loss[i]
  const float offIt = (1.f / (NN - 1.f)) * expf(-(ARR(8)[i] + nbI) * INV_T);
  const float denomIt = fmaxf(sI / ((float)BSZ - 1.f) + offIt, EPS_CLAMP);
  ARR(13)[i] = 1.f / denomIt;  // recipI_tilde
}

// ---------------------------------------------------------------------------
// Kernel 7: column pass -> g_T[j], num_txt[j], colsum of w_img_tilde
// ---------------------------------------------------------------------------
__global__ void k_colpass(const float* __restrict__ sim, float* __restrict__ ws) {
  __shared__ float s1[256], s2[256], s3[256];
  const int tc = threadIdx.x & 63, tr = threadIdx.x >> 6;
  const int j = blockIdx.x * 64 + tc;
  const float* diag = ARR(0);
  const float* rt = ARR(9);  // diag[i] + nb_I[i]
  const float* zIb = ARR(7);
  const float* recipI = ARR(13);
  const float dj = diag[j];
  const float ct = dj + ARR(4)[j];  // diag[j] + nb_T[j]
  const float zTbj = ARR(8)[j];
  float gT = 0.f, numT = 0.f, wsum = 0.f;
  for (int i = tr; i < BSZ; i += 4) {
    const float s = sim[(size_t)i * BSZ + j];
    const float et = expf((s - ct - zIb[i]) * INV_T);
    if (i != j) {
      gT += et;
      numT += et * (s - dj);
    }
    const float ei = expf((s - rt[i] - zTbj) * INV_T);
    wsum += ei * recipI[i];
  }
  s1[threadIdx.x] = gT;
  s2[threadIdx.x] = numT;
  s3[threadIdx.x] = wsum;
  __syncthreads();
  if (tr == 0) {
    ARR(15)[j] = s1[tc] + s1[tc + 64] + s1[tc + 128] + s1[tc + 192];
    ARR(16)[j] = s2[tc] + s2[tc + 64] + s2[tc + 128] + s2[tc + 192];
    ARR(19)[j] = s3[tc] + s3[tc + 64] + s3[tc + 128] + s3[tc + 192];
  }
}

// ---------------------------------------------------------------------------
// Kernel 8: s_T update, text_loss, 1/denomT_tilde, tgt_T, zetaT_new
// ---------------------------------------------------------------------------
__global__ void k_smallT(const float* __restrict__ s_T, const float* __restrict__ z_T,
                         const int* __restrict__ idT, float* __restrict__ ws) {
  const int j = blockIdx.x * 256 + threadIdx.x;
  if (j >= BSZ) return;
  const float g = ARR(15)[j];
  const float obT = ARR(6)[j], nbT = ARR(4)[j];
  float s0 = s_T[idT[j]];
  if (s0 == 0.f) s0 = g * expf(nbT - obT);
  const float sT = (1.f - GAMMA) * s0 * expf(obT - nbT) + GAMMA * g;
  const float offT = ((float)BSZ - 1.f) / (NN - 1.f) * expf(-nbT * INV_T);
  const float denom = fmaxf(sT + offT, EPS_CLAMP);
  ARR(18)[j] = ARR(16)[j] / denom;  // text_loss[j]
  const float offTt = (1.f / (NN - 1.f)) * expf(-(ARR(7)[j] + nbT) * INV_T);
  const float denomTt = fmaxf(sT / ((float)BSZ - 1.f) + offTt, EPS_CLAMP);
  ARR(17)[j] = 1.f / denomTt;  // recipT_tilde
  const float tgtT = -(NN / (NN - 1.f)) * (ARR(19)[j] / (float)BSZ) + 1.f;
  float z0 = z_T[idT[j]];
  if (z0 == 0.f) z0 = tgtT;
  const float zTn = (1.f - THETA) * z0 + THETA * tgtT;
  const float eta = ARR(23)[0];
  ARR(22)[j] = ARR(8)[j] - eta * zTn;  // zetaT_b_new
}

// ---------------------------------------------------------------------------
// Kernel 9: second row pass -> rowsum of w_txt_tilde
// ---------------------------------------------------------------------------
__global__ void k_rowpass2(const float* __restrict__ sim, float* __restrict__ ws) {
  __shared__ float sm[256];
  const int i = blockIdx.x;
  const float* row = sim + (size_t)i * BSZ;
  const float* ct = ARR(10);  // diag[j] + nb_T[j]
  const float* recipT = ARR(17);
  const float zi = ARR(7)[i];
  float acc = 0.f;
  for (int j = threadIdx.x; j < BSZ; j += 256)
    acc += expf((row[j] - ct[j] - zi) * INV_T) * recipT[j];
  sm[threadIdx.x] = acc;
  __syncthreads();
  for (int s = 128; s > 0; s >>= 1) {
    if (threadIdx.x < s) sm[threadIdx.x] += sm[threadIdx.x + s];
    __syncthreads();
  }
  if (threadIdx.x == 0) ARR(20)[i] = sm[0];
}

// ---------------------------------------------------------------------------
// Kernel 10: tgt_I, zetaI_new
// ---------------------------------------------------------------------------
__global__ void k_smallI2(const float* __restrict__ z_I, const int* __restrict__ idI,
                          float* __restrict__ ws) {
  const int i = blockIdx.x * 256 + threadIdx.x;
  if (i >= BSZ) return;
  const float tgtI = -(NN / (NN - 1.f)) * (ARR(20)[i] / (float)BSZ) + 1.f;
  float z0 = z_I[idI[i]];
  if (z0 == 0.f) z0 = tgtI;
  const float zIn = (1.f - THETA) * z0 + THETA * tgtI;
  const float eta = ARR(23)[0];  // ETA_I_RATIO == 1
  ARR(21)[i] = ARR(7)[i] - eta * zIn;  // zetaI_b_new
}

// ---------------------------------------------------------------------------
// Kernel 11: final reductions -> 9 output scalars
// ---------------------------------------------------------------------------
__global__ void k_final(float* __restrict__ ws, float* __restrict__ out) {
  __shared__ float sm[8][256];
  const int t = threadIdx.x;
  float sIL = 0.f, sTL = 0.f, sZI = 0.f, sZT = 0.f;
  float mxZI = -INFINITY, mnZI = INFINITY, mxZT = -INFINITY, mnZT = INFINITY;
  for (int i = t; i < BSZ; i += 256) {
    sIL += ARR(14)[i];
    sTL += ARR(18)[i];
    const float zi = ARR(21)[i], zt = ARR(22)[i];
    sZI += zi;
    sZT += zt;
    mxZI = fmaxf(mxZI, zi);
    mnZI = fminf(mnZI, zi);
    mxZT = fmaxf(mxZT, zt);
    mnZT = fminf(mnZT, zt);
  }
  sm[0][t] = sIL;
  sm[1][t] = sTL;
  sm[2][t] = sZI;
  sm[3][t] = sZT;
  sm[4][t] = mxZI;
  sm[5][t] = mnZI;
  sm[6][t] = mxZT;
  sm[7][t] = mnZT;
  __syncthreads();
  for (int s = 128; s > 0; s >>= 1) {
    if (t < s) {
      sm[0][t] += sm[0][t + s];
      sm[1][t] += sm[1][t + s];
      sm[2][t] += sm[2][t + s];
      sm[3][t] += sm[3][t + s];
      sm[4][t] = fmaxf(sm[4][t], sm[4][t + s]);
      sm[5][t] = fminf(sm[5][t], sm[5][t + s]);
      sm[6][t] = fmaxf(sm[6][t], sm[6][t + s]);
      sm[7][t] = fminf(sm[7][t], sm[7][t + s]);
    }
    __syncthreads();
  }
  if (t == 0) {
    const float eta = ARR(23)[0];
    out[0] = sm[0][0] / (float)BSZ + sm[1][0] / (float)BSZ;  // total_loss
    out[1] = sm[4][0];                                       // max zetaI
    out[2] = sm[2][0] / (float)BSZ;                          // mean zetaI
    out[3] = sm[5][0];                                       // min zetaI
    out[4] = sm[6][0];                                       // max zetaT
    out[5] = sm[3][0] / (float)BSZ;                          // mean zetaT
    out[6] = sm[7][0];                                       // min zetaT
    out[7] = eta;                                            // cur_eta_I
    out[8] = eta;                                            // cur_eta_T
  }
}

extern "C" void kernel_launch(void* const* d_in, const int* in_sizes, int n_in,
                              void* d_out, int out_size, void* d_ws, size_t ws_size,
                              hipStream_t stream) {
  (void)in_sizes; (void)n_in; (void)out_size; (void)ws_size;
  const float* img = (const float*)d_in[0];
  const float* txt = (const float*)d_in[1];
  const int* idI = (const int*)d_in[2];
  const int* idT = (const int*)d_in[3];
  const float* s_I = (const float*)d_in[4];
  const float* s_T = (const float*)d_in[5];
  const float* b_I = (const float*)d_in[6];
  const float* b_T = (const float*)d_in[7];
  const float* z_I = (const float*)d_in[8];
  const float* z_T = (const float*)d_in[9];
  const float* zeta_I = (const float*)d_in[10];
  const float* zeta_T = (const float*)d_in[11];
  const int* epoch = (const int*)d_in[12];
  const int* max_epoch = (const int*)d_in[13];
  float* out = (float*)d_out;
  float* ws = (float*)d_ws;
  float* sim = ws;

  k_gemm<<<dim3(BSZ / TN, BSZ / TM), 256, 0, stream>>>(img, txt, sim);
  k_rowmax<<<BSZ, 256, 0, stream>>>(sim, ws);
  k_colmax<<<BSZ / 64, 256, 0, stream>>>(sim, ws);
  k_prep<<<BSZ / 256, 256, 0, stream>>>(idI, idT, b_I, b_T, zeta_I, zeta_T,
                                        epoch, max_epoch, ws);
  k_rowpass1<<<BSZ, 256, 0, stream>>>(sim, ws);
  k_smallI<<<BSZ / 256, 256, 0, stream>>>(s_I, idI, ws);
  k_colpass<<<BSZ / 64, 256, 0, stream>>>(sim, ws);
  k_smallT<<<BSZ / 256, 256, 0, stream>>>(s_T, z_T, idT, ws);
  k_rowpass2<<<BSZ, 256, 0, stream>>>(sim, ws);
  k_smallI2<<<BSZ / 256, 256, 0, stream>>>(z_I, idI, ws);
  k_final<<<1, 256, 0, stream>>>(ws, out);
}